// hex_pooling_max_32968168964589
// MI455X (gfx1250) — compile-verified
//
#include <hip/hip_runtime.h>
#include <stdint.h>

#define FEAT 64
#define HEX 7
#define NODES_PER_BLOCK 4
#define BLOCK_THREADS (NODES_PER_BLOCK * 64)

// Address-space typed pointers for the CDNA5 async global->LDS builtin
// (ROCm clang declares the params as int* in AS1 / AS3).
typedef __attribute__((address_space(1))) int g_int_t;
typedef __attribute__((address_space(3))) int l_int_t;
typedef __attribute__((address_space(3))) void lds_void_t;

#if defined(__gfx1250__) && __has_builtin(__builtin_amdgcn_global_load_async_to_lds_b32)
#define HEXPOOL_ASYNC_BUILTIN 1
#elif defined(__gfx1250__)
#define HEXPOOL_ASYNC_ASM 1
#endif

__device__ __forceinline__ void async_gather_b32(const float* src, float* lds_dst) {
#if defined(HEXPOOL_ASYNC_BUILTIN)
    // (global src, lds dst, imm offset, imm cpol)
    __builtin_amdgcn_global_load_async_to_lds_b32(
        (g_int_t*)src, (l_int_t*)lds_dst, 0, 0);
#elif defined(HEXPOOL_ASYNC_ASM)
    unsigned lds_off = (unsigned)(uintptr_t)(lds_void_t*)lds_dst;
    asm volatile("global_load_async_to_lds_b32 %0, %1, off"
                 :: "v"(lds_off), "v"(src)
                 : "memory");
#else
    *lds_dst = *src;   // host pass / non-gfx1250 fallback
#endif
}

__device__ __forceinline__ void wait_async_done() {
#if defined(HEXPOOL_ASYNC_BUILTIN) || defined(HEXPOOL_ASYNC_ASM)
#if __has_builtin(__builtin_amdgcn_s_wait_asynccnt)
    __builtin_amdgcn_s_wait_asynccnt(0);
#else
    asm volatile("s_wait_asynccnt 0" ::: "memory");
#endif
#endif
}

// out[n, f] = max_{j=0..6} flat[f*7 + j], flat[k] = x[hex_idx[n, k>>6]][k&63]
__global__ __launch_bounds__(BLOCK_THREADS)
void hex_pooling_max_kernel(const float* __restrict__ x,
                            const int* __restrict__ hex_idx,
                            float* __restrict__ out,
                            int num_nodes) {
    __shared__ float lds[NODES_PER_BLOCK * HEX * FEAT];

    const int tid   = threadIdx.x;
    const int local = tid >> 6;          // node slot within block (0..3)
    const int f     = tid & 63;          // feature lane (0..63)
    const int node  = blockIdx.x * NODES_PER_BLOCK + local;

    float* node_lds = &lds[local * (HEX * FEAT)];

    if (node < num_nodes) {
        const int* idx = hex_idx + node * HEX;
        // 7 coalesced 256B row gathers, streamed straight into LDS
        // (lds holds exactly the reference's flattened 448-float block).
        #pragma unroll
        for (int j = 0; j < HEX; ++j) {
            const float* src = x + (size_t)idx[j] * FEAT + f;
            async_gather_b32(src, node_lds + j * FEAT + f);
        }
    }
    wait_async_done();
    __syncthreads();

    if (node < num_nodes) {
        // Windowed max over 7 consecutive flat elements; stride 7 is
        // coprime with 64 LDS banks -> conflict-free across the wave.
        const float* w = node_lds + f * HEX;
        float m = w[0];
        #pragma unroll
        for (int j = 1; j < HEX; ++j) m = fmaxf(m, w[j]);
        out[(size_t)node * FEAT + f] = m;
    }
}

extern "C" void kernel_launch(void* const* d_in, const int* in_sizes, int n_in,
                              void* d_out, int out_size, void* d_ws, size_t ws_size,
                              hipStream_t stream) {
    const float* x       = (const float*)d_in[0];
    const int*   hex_idx = (const int*)d_in[1];   // int64 in source, int32 under JAX default x64-off
    float*       out     = (float*)d_out;

    const int num_nodes = out_size / FEAT;        // 163842
    const int blocks    = (num_nodes + NODES_PER_BLOCK - 1) / NODES_PER_BLOCK;

    hex_pooling_max_kernel<<<blocks, BLOCK_THREADS, 0, stream>>>(x, hex_idx, out, num_nodes);
}